// SpearmanCorrLoss_14723147891027
// MI455X (gfx1250) — compile-verified
//
#include <hip/hip_runtime.h>
#include <math.h>

// Spearman soft-rank correlation loss, N = 262144 (2^18), REG_STRENGTH = 1.0
// Pipeline: descending bitonic argsort (LDS-fused) -> two-level PAV isotonic
// regression -> scatter ranks -> two-pass Pearson corr with WMMA f32 folds.

#define CHUNK   256     // elements per parallel-PAV chunk
#define NWAVES  128     // waves for the WMMA reduction kernels

typedef __attribute__((ext_vector_type(2))) float v2f;
typedef __attribute__((ext_vector_type(8))) float v8f;

// strict total order: "a ranks before b" in descending sort, ties by ascending index
__device__ __forceinline__ bool before_desc(float ka, int ia, float kb, int ib) {
    return (ka > kb) || (ka == kb && ia < ib);
}

// ---------------------------------------------------------------- init
__global__ void init_keys(const float* __restrict__ pred, const float* __restrict__ target,
                          float* keyP, int* idxP, float* keyT, int* idxT, int n) {
    int i = blockIdx.x * blockDim.x + threadIdx.x;
    if (i < n) {
        keyP[i] = pred[i];     // theta = x * (1/REG), REG = 1
        keyT[i] = target[i];
        idxP[i] = i;
        idxT[i] = i;
    }
}

// ---------------------------------------------------------------- bitonic sort
// Full local sort of 1024-element tiles in LDS: covers merge sizes k = 2..1024.
__global__ void bitonic_local(float* key, int* idx) {
    __shared__ float sk[1024];
    __shared__ int   si[1024];
    const int base = blockIdx.x * 1024;
    for (int t = threadIdx.x; t < 1024; t += blockDim.x) { sk[t] = key[base + t]; si[t] = idx[base + t]; }
    __syncthreads();
    for (int k = 2; k <= 1024; k <<= 1) {
        for (int j = k >> 1; j >= 1; j >>= 1) {
            const int t  = threadIdx.x;                       // 512 threads -> 512 pairs
            const int i1 = ((t & ~(j - 1)) << 1) | (t & (j - 1));
            const int i2 = i1 | j;
            const bool dir = (((base + i1) & k) == 0);        // true => descending segment
            float k1 = sk[i1], k2 = sk[i2];
            int   a1 = si[i1], a2 = si[i2];
            bool  b  = before_desc(k1, a1, k2, a2);
            if (b != dir) { sk[i1] = k2; si[i1] = a2; sk[i2] = k1; si[i2] = a1; }
            __syncthreads();
        }
    }
    for (int t = threadIdx.x; t < 1024; t += blockDim.x) { key[base + t] = sk[t]; idx[base + t] = si[t]; }
}

// One global compare-exchange pass (used only for j >= 1024).
__global__ void bitonic_global(float* key, int* idx, int j, int k) {
    const int t  = blockIdx.x * blockDim.x + threadIdx.x;     // n/2 threads
    const int i1 = ((t & ~(j - 1)) << 1) | (t & (j - 1));
    const int i2 = i1 | j;
    const bool dir = ((i1 & k) == 0);
    float k1 = key[i1], k2 = key[i2];
    int   a1 = idx[i1], a2 = idx[i2];
    bool  b  = before_desc(k1, a1, k2, a2);
    if (b != dir) { key[i1] = k2; idx[i1] = a2; key[i2] = k1; idx[i2] = a1; }
}

// Fused tail of a merge stage: j = 512..1 entirely in LDS.
__global__ void bitonic_tail(float* key, int* idx, int k) {
    __shared__ float sk[1024];
    __shared__ int   si[1024];
    const int base = blockIdx.x * 1024;
    for (int t = threadIdx.x; t < 1024; t += blockDim.x) { sk[t] = key[base + t]; si[t] = idx[base + t]; }
    __syncthreads();
    for (int j = 512; j >= 1; j >>= 1) {
        const int t  = threadIdx.x;
        const int i1 = ((t & ~(j - 1)) << 1) | (t & (j - 1));
        const int i2 = i1 | j;
        const bool dir = (((base + i1) & k) == 0);
        float k1 = sk[i1], k2 = sk[i2];
        int   a1 = si[i1], a2 = si[i2];
        bool  b  = before_desc(k1, a1, k2, a2);
        if (b != dir) { sk[i1] = k2; si[i1] = a2; sk[i2] = k1; si[i2] = a1; }
        __syncthreads();
    }
    for (int t = threadIdx.x; t < 1024; t += blockDim.x) { key[base + t] = sk[t]; idx[base + t] = si[t]; }
}

// ---------------------------------------------------------------- PAV (two-level)
// Level 1: each thread PAV-pools one 256-element chunk of y = s[i] - (n - i).
// Adjacent-violator merges are locally valid, so chunk pooling is globally correct.
__global__ void pav_chunk(const float* __restrict__ key, float* bsum, float* bcnt, int* bnum, int n) {
    const int c = blockIdx.x * blockDim.x + threadIdx.x;
    if (c >= n / CHUNK) return;
    const int base = c * CHUNK;
    float* S  = bsum + base;
    float* Cn = bcnt + base;
    int p = 0;
    for (int t = 0; t < CHUNK; ++t) {
        const int i = base + t;
        const float y = key[i] - (float)(n - i);
        float s = y, cc = 1.0f, m = y;
        while (p >= 1) {
            const float ps = S[p - 1], pc = Cn[p - 1];
            if (m > ps / pc) { s += ps; cc += pc; m = s / cc; --p; }
            else break;
        }
        S[p] = s; Cn[p] = cc; ++p;
    }
    bnum[c] = p;
}

// Level 2: sequential block-level PAV over the surviving blocks, compacted
// in place (write index provably <= read index). Then convert the final
// stack into (mean[b], cumulative-count[b]) in place.
__global__ void pav_merge(float* bsumP, float* bcntP, const int* bnumP, int* fpP,
                          float* bsumT, float* bcntT, const int* bnumT, int* fpT, int nchunk) {
    if (threadIdx.x != 0) return;
    float* bsum; float* bcnt; const int* bnum; int* fp;
    if (blockIdx.x == 0) { bsum = bsumP; bcnt = bcntP; bnum = bnumP; fp = fpP; }
    else                 { bsum = bsumT; bcnt = bcntT; bnum = bnumT; fp = fpT; }
    int p = 0;
    for (int c = 0; c < nchunk; ++c) {
        const int nb = bnum[c];
        const int rb = c * CHUNK;
        for (int t = 0; t < nb; ++t) {
            float s = bsum[rb + t], cc = bcnt[rb + t];
            float m = s / cc;
            while (p >= 1) {
                const float ps = bsum[p - 1], pc = bcnt[p - 1];
                if (m > ps / pc) { s += ps; cc += pc; m = s / cc; --p; }
                else break;
            }
            bsum[p] = s; bcnt[p] = cc; ++p;
        }
    }
    float run = 0.0f;
    for (int b = 0; b < p; ++b) {
        const float m = bsum[b] / bcnt[b];
        run += bcnt[b];
        bsum[b] = m;      // block mean
        bcnt[b] = run;    // cumulative count (exact in f32 up to 2^24)
    }
    *fp = p;
}

// Expand block solution + scatter: rank[idx[i]] = s[i] - mean[block(i)],
// block(i) found by searchsorted(cum, i, 'right') as in the reference.
__global__ void expand_scatter(const float* __restrict__ key, const int* __restrict__ idx,
                               const float* __restrict__ mean, const float* __restrict__ cum,
                               const int* __restrict__ fp, float* rank, int n) {
    const int i = blockIdx.x * blockDim.x + threadIdx.x;
    if (i >= n) return;
    const int P = *fp;
    int lo = 0, hi = P - 1;
    const float fi = (float)i;
    while (lo < hi) {
        const int mid = (lo + hi) >> 1;
        if (cum[mid] > fi) hi = mid; else lo = mid + 1;
    }
    rank[idx[i]] = key[i] - mean[lo];
}

// ---------------------------------------------------------------- WMMA reductions
// V_WMMA_F32_16X16X4_F32 with B = all-ones: D[m][n] += sum_k A[m][k], i.e. a
// 4-way fold per lane per instruction on the matrix pipe. Summing all 8 C
// registers over all 32 lanes yields 16x the true sum (16 identical columns).
// A-tile mapping for a row-major 64-element chunk (e = m*4 + k):
//   lane l<16 : VGPRs hold e = 4l, 4l+1 ; lane l>=16 : e = 4(l-16)+2, +3.
__device__ __forceinline__ int a_tile_off(int l) { return 4 * (l & 15) + 2 * (l >> 4); }

__global__ void mean_partial_wmma(const float* __restrict__ rp, const float* __restrict__ rt,
                                  float* px, float* py, int n) {
    const int w = blockIdx.x, l = threadIdx.x;
    const int span = n / gridDim.x;            // 2048
    const int base = w * span;
    const int sub  = a_tile_off(l);
    v8f ax = {0.f,0.f,0.f,0.f,0.f,0.f,0.f,0.f};
    v8f ay = ax;
    const v2f ones = {1.0f, 1.0f};
    for (int o = 0; o < span; o += 64) {
        v2f a = *(const v2f*)(rp + base + o + sub);
        v2f b = *(const v2f*)(rt + base + o + sub);
        ax = __builtin_amdgcn_wmma_f32_16x16x4_f32(false, a, false, ones, (short)0, ax, false, false);
        ay = __builtin_amdgcn_wmma_f32_16x16x4_f32(false, b, false, ones, (short)0, ay, false, false);
    }
    float sx = 0.f, sy = 0.f;
    for (int i = 0; i < 8; ++i) { sx += ax[i]; sy += ay[i]; }
    for (int off = 16; off > 0; off >>= 1) { sx += __shfl_xor(sx, off); sy += __shfl_xor(sy, off); }
    if (l == 0) { px[w] = sx * (1.0f / 16.0f); py[w] = sy * (1.0f / 16.0f); }
}

__global__ void mean_finalize(const float* px, const float* py, double* means, int nw, int n) {
    __shared__ double sx[128], sy[128];
    const int t = threadIdx.x;
    sx[t] = (t < nw) ? (double)px[t] : 0.0;
    sy[t] = (t < nw) ? (double)py[t] : 0.0;
    __syncthreads();
    for (int s = 64; s > 0; s >>= 1) {
        if (t < s) { sx[t] += sx[t + s]; sy[t] += sy[t + s]; }
        __syncthreads();
    }
    if (t == 0) { means[0] = sx[0] / (double)n; means[1] = sy[0] / (double)n; }
}

__global__ void cov_partial_wmma(const float* __restrict__ rp, const float* __restrict__ rt,
                                 const double* __restrict__ means,
                                 float* pxy, float* pxx, float* pyy, int n) {
    const float mx = (float)means[0], my = (float)means[1];
    const int w = blockIdx.x, l = threadIdx.x;
    const int span = n / gridDim.x;
    const int base = w * span;
    const int sub  = a_tile_off(l);
    v8f axy = {0.f,0.f,0.f,0.f,0.f,0.f,0.f,0.f};
    v8f axx = axy, ayy = axy;
    const v2f ones = {1.0f, 1.0f};
    for (int o = 0; o < span; o += 64) {
        v2f a = *(const v2f*)(rp + base + o + sub);
        v2f b = *(const v2f*)(rt + base + o + sub);
        a -= mx; b -= my;
        v2f ab = a * b, aa = a * a, bb = b * b;
        axy = __builtin_amdgcn_wmma_f32_16x16x4_f32(false, ab, false, ones, (short)0, axy, false, false);
        axx = __builtin_amdgcn_wmma_f32_16x16x4_f32(false, aa, false, ones, (short)0, axx, false, false);
        ayy = __builtin_amdgcn_wmma_f32_16x16x4_f32(false, bb, false, ones, (short)0, ayy, false, false);
    }
    float sxy = 0.f, sxx = 0.f, syy = 0.f;
    for (int i = 0; i < 8; ++i) { sxy += axy[i]; sxx += axx[i]; syy += ayy[i]; }
    for (int off = 16; off > 0; off >>= 1) {
        sxy += __shfl_xor(sxy, off); sxx += __shfl_xor(sxx, off); syy += __shfl_xor(syy, off);
    }
    if (l == 0) {
        pxy[w] = sxy * (1.0f / 16.0f);
        pxx[w] = sxx * (1.0f / 16.0f);
        pyy[w] = syy * (1.0f / 16.0f);
    }
}

__global__ void corr_finalize(const float* pxy, const float* pxx, const float* pyy,
                              float* out, int nw) {
    __shared__ double s0[128], s1[128], s2[128];
    const int t = threadIdx.x;
    s0[t] = (t < nw) ? (double)pxy[t] : 0.0;
    s1[t] = (t < nw) ? (double)pxx[t] : 0.0;
    s2[t] = (t < nw) ? (double)pyy[t] : 0.0;
    __syncthreads();
    for (int s = 64; s > 0; s >>= 1) {
        if (t < s) { s0[t] += s0[t + s]; s1[t] += s1[t + s]; s2[t] += s2[t + s]; }
        __syncthreads();
    }
    if (t == 0) {
        const double corr = s0[0] / sqrt(s1[0] * s2[0]);
        out[0] = (float)(-corr);
    }
}

// ---------------------------------------------------------------- launcher
extern "C" void kernel_launch(void* const* d_in, const int* in_sizes, int n_in,
                              void* d_out, int out_size, void* d_ws, size_t ws_size,
                              hipStream_t stream) {
    (void)n_in; (void)out_size; (void)ws_size;
    const int n = in_sizes[0];                 // 262144, power of two
    const int nchunk = n / CHUNK;
    const float* pred   = (const float*)d_in[0];
    const float* target = (const float*)d_in[1];
    float* out = (float*)d_out;

    // workspace layout: 8N floats + 2N ints + misc (~10.5 MB total)
    float* f     = (float*)d_ws;
    float* keyP  = f;
    float* keyT  = f + (size_t)n;
    float* bsumP = f + 2 * (size_t)n;   // -> block means after merge
    float* bcntP = f + 3 * (size_t)n;   // -> cumulative counts after merge
    float* bsumT = f + 4 * (size_t)n;
    float* bcntT = f + 5 * (size_t)n;
    float* rankP = f + 6 * (size_t)n;
    float* rankT = f + 7 * (size_t)n;
    int*   ip    = (int*)(f + 8 * (size_t)n);
    int*   idxP  = ip;
    int*   idxT  = ip + (size_t)n;
    double* means = (double*)(ip + 2 * (size_t)n);   // 8-byte aligned (n even)
    float* px  = (float*)(means + 2);
    float* py  = px  + NWAVES;
    float* pxy = py  + NWAVES;
    float* pxx = pxy + NWAVES;
    float* pyy = pxx + NWAVES;
    int* bnumP = (int*)(pyy + NWAVES);
    int* bnumT = bnumP + nchunk;
    int* fpP   = bnumT + nchunk;
    int* fpT   = fpP + 1;

    init_keys<<<n / 256, 256, 0, stream>>>(pred, target, keyP, idxP, keyT, idxT, n);

    float* keys[2] = {keyP, keyT};
    int*   idxs[2] = {idxP, idxT};
    for (int a = 0; a < 2; ++a) {
        bitonic_local<<<n / 1024, 512, 0, stream>>>(keys[a], idxs[a]);
        for (int k = 2048; k <= n; k <<= 1) {
            for (int j = k >> 1; j >= 1024; j >>= 1)
                bitonic_global<<<(n / 2) / 256, 256, 0, stream>>>(keys[a], idxs[a], j, k);
            bitonic_tail<<<n / 1024, 512, 0, stream>>>(keys[a], idxs[a], k);
        }
    }

    pav_chunk<<<(nchunk + 255) / 256, 256, 0, stream>>>(keyP, bsumP, bcntP, bnumP, n);
    pav_chunk<<<(nchunk + 255) / 256, 256, 0, stream>>>(keyT, bsumT, bcntT, bnumT, n);
    pav_merge<<<2, 32, 0, stream>>>(bsumP, bcntP, bnumP, fpP, bsumT, bcntT, bnumT, fpT, nchunk);
    expand_scatter<<<n / 256, 256, 0, stream>>>(keyP, idxP, bsumP, bcntP, fpP, rankP, n);
    expand_scatter<<<n / 256, 256, 0, stream>>>(keyT, idxT, bsumT, bcntT, fpT, rankT, n);

    mean_partial_wmma<<<NWAVES, 32, 0, stream>>>(rankP, rankT, px, py, n);
    mean_finalize<<<1, 128, 0, stream>>>(px, py, means, NWAVES, n);
    cov_partial_wmma<<<NWAVES, 32, 0, stream>>>(rankP, rankT, means, pxy, pxx, pyy, n);
    corr_finalize<<<1, 128, 0, stream>>>(pxy, pxx, pyy, out, NWAVES);
}